// GraphAttentionLayer_36962488549475
// MI455X (gfx1250) — compile-verified
//
#include <hip/hip_runtime.h>

#define B_ 4
#define N_ 2048
#define D_ 128
#define H_ 4
#define NT_ (N_/16)      // 128 row tiles
#define JC_ 128          // j-chunk (K chunk for the att@ht GEMM)
#define ESTRIDE 132      // padded LDS row stride (floats), 16B-aligned rows
#define ALPHA_ 0.2f

typedef float v2f __attribute__((ext_vector_type(2)));
typedef float v8f __attribute__((ext_vector_type(8)));

__device__ __forceinline__ float wave_sum32(float v) {
#pragma unroll
    for (int m = 1; m < 32; m <<= 1) v += __shfl_xor(v, m, 32);
    return v;
}

// ---------------------------------------------------------------------------
// Kernel 1: ht[b,h,n,:] = x[b,n,:] @ W[h] + Wb[h]; s1/s2 = ht . a1/a2 + bias
// One wave per (b,h,ntile): 16x128 output tile via V_WMMA_F32_16X16X4_F32.
// ---------------------------------------------------------------------------
__global__ __launch_bounds__(32) void k_ht(
    const float* __restrict__ x, const float* __restrict__ W,
    const float* __restrict__ Wb, const float* __restrict__ a1,
    const float* __restrict__ a1b, const float* __restrict__ a2,
    const float* __restrict__ a2b,
    float* __restrict__ ht, float* __restrict__ s1, float* __restrict__ s2)
{
    const int blk = blockIdx.x;
    const int nt = blk % NT_;
    const int bh = blk / NT_;
    const int hd = bh % H_;
    const int b  = bh / H_;
    const int lane = threadIdx.x;
    const int lo = lane & 15, hi = lane >> 4;

    // A fragment source: row m = lo of x-tile; k = k0 + 2*hi + {0,1}
    const float* xp = x + ((size_t)b * N_ + (size_t)(nt * 16 + lo)) * D_;
    // B fragment source: W[h][k][d], d-column base = lo
    const float* wp = W + (size_t)hd * D_ * D_ + lo;

    v8f acc[8];
#pragma unroll
    for (int dt = 0; dt < 8; ++dt)
#pragma unroll
        for (int v = 0; v < 8; ++v) acc[dt][v] = 0.f;

#pragma unroll 2
    for (int k0 = 0; k0 < D_; k0 += 4) {
        const int ka = k0 + 2 * hi;
        v2f afrag = *(const v2f*)(xp + ka);
        const float* wr = wp + (size_t)ka * D_;
#pragma unroll
        for (int dt = 0; dt < 8; ++dt) {
            v2f bfrag;
            bfrag.x = wr[dt * 16];
            bfrag.y = wr[dt * 16 + D_];
            acc[dt] = __builtin_amdgcn_wmma_f32_16x16x4_f32(
                false, afrag, false, bfrag, (short)0, acc[dt], false, false);
        }
    }

    float wbv[8], a1v[8], a2v[8];
#pragma unroll
    for (int dt = 0; dt < 8; ++dt) {
        wbv[dt] = Wb[hd * D_ + dt * 16 + lo];
        a1v[dt] = a1[hd * D_ + dt * 16 + lo];
        a2v[dt] = a2[hd * D_ + dt * 16 + lo];
    }

    float* htp = ht + ((size_t)(b * H_ + hd) * N_ + (size_t)nt * 16) * D_;
    float p1[8], p2[8];
#pragma unroll
    for (int v = 0; v < 8; ++v) { p1[v] = 0.f; p2[v] = 0.f; }

#pragma unroll
    for (int dt = 0; dt < 8; ++dt) {
#pragma unroll
        for (int v = 0; v < 8; ++v) {
            float val = acc[dt][v] + wbv[dt];        // +Wb
            p1[v] += val * a1v[dt];
            p2[v] += val * a2v[dt];
            // D layout: row = v + 8*hi, col = dt*16 + lo  (coalesced over lo)
            htp[(size_t)(v + 8 * hi) * D_ + dt * 16 + lo] = val;
        }
    }
    // reduce dots across the 16 lanes of each half-wave
#pragma unroll
    for (int v = 0; v < 8; ++v) {
#pragma unroll
        for (int m = 1; m < 16; m <<= 1) {
            p1[v] += __shfl_xor(p1[v], m, 32);
            p2[v] += __shfl_xor(p2[v], m, 32);
        }
    }
    if (lo == 0) {
#pragma unroll
        for (int v = 0; v < 8; ++v) {
            int row = nt * 16 + v + 8 * hi;
            s1[(size_t)(b * H_ + hd) * N_ + row] = p1[v] + a1b[hd];
            s2[(size_t)(b * H_ + hd) * N_ + row] = p2[v] + a2b[hd];
        }
    }
}

// ---------------------------------------------------------------------------
// Kernel 2: per-(b,h) max over s2 (LeakyReLU is monotone => row max of scores
// is LR(s1_i + max_j s2_j)).  16 waves, one per (b,h).
// ---------------------------------------------------------------------------
__global__ __launch_bounds__(512) void k_s2max(const float* __restrict__ s2,
                                               float* __restrict__ s2max)
{
    const int w = threadIdx.x >> 5;    // 0..15 == b*H+h
    const int lane = threadIdx.x & 31;
    const float* p = s2 + (size_t)w * N_;
    float mx = -3.402823466e38f;
    for (int j = lane; j < N_; j += 32) mx = fmaxf(mx, p[j]);
#pragma unroll
    for (int m = 1; m < 32; m <<= 1) mx = fmaxf(mx, __shfl_xor(mx, m, 32));
    if (lane == 0) s2max[w] = mx;
}

// ---------------------------------------------------------------------------
// Kernel 3: attention + hp = att@ht (WMMA f32) + fused head-mean/tanh output.
// Block = (b, ntile of 16 rows), 4 waves, wave = head.
// ---------------------------------------------------------------------------
__global__ __launch_bounds__(128) void k_att(
    const float* __restrict__ adj, const float* __restrict__ x,
    const float* __restrict__ ht, const float* __restrict__ s1,
    const float* __restrict__ s2, const float* __restrict__ s2max,
    float* __restrict__ out, float* __restrict__ att)
{
    // e-chunk staging (4 waves x 16 x ESTRIDE) ; later reused as hp buffer
    // (4 waves x 16 x 128).  8448 floats = 33 KB.
    __shared__ float smem[H_ * 16 * ESTRIDE];

    const int b  = blockIdx.x / NT_;
    const int nt = blockIdx.x % NT_;
    const int hd = threadIdx.x >> 5;
    const int lane = threadIdx.x & 31;
    const int lo = lane & 15, hi = lane >> 4;
    const int n0 = nt * 16;

    const float* adjB = adj + ((size_t)b * N_ + n0) * N_;
    const float* htp  = ht  + (size_t)(b * H_ + hd) * N_ * D_;
    const float* s1p  = s1  + (size_t)(b * H_ + hd) * N_ + n0;
    const float* s2p  = s2  + (size_t)(b * H_ + hd) * N_;
    const float  s2m  = s2max[b * H_ + hd];

    float s1row[16], mrow[16];
#pragma unroll
    for (int i = 0; i < 16; ++i) {
        float sv = s1p[i];
        s1row[i] = sv;
        float sc = sv + s2m;
        mrow[i] = sc > 0.f ? sc : ALPHA_ * sc;   // exact row max of scores
    }

    // ---- pass 1: denominators ----
    float dp[16];
#pragma unroll
    for (int i = 0; i < 16; ++i) dp[i] = 0.f;

#pragma unroll 1
    for (int j0 = 0; j0 < N_; j0 += JC_) {
        float4 s2v = *(const float4*)(s2p + j0 + 4 * lane);
#pragma unroll
        for (int i = 0; i < 16; ++i) {
            float4 av = *(const float4*)(adjB + (size_t)i * N_ + j0 + 4 * lane);
            float c0 = s1row[i] + s2v.x; c0 = c0 > 0.f ? c0 : ALPHA_ * c0;
            float c1 = s1row[i] + s2v.y; c1 = c1 > 0.f ? c1 : ALPHA_ * c1;
            float c2 = s1row[i] + s2v.z; c2 = c2 > 0.f ? c2 : ALPHA_ * c2;
            float c3 = s1row[i] + s2v.w; c3 = c3 > 0.f ? c3 : ALPHA_ * c3;
            float e0 = __expf(c0 - mrow[i]) * av.x;
            float e1 = __expf(c1 - mrow[i]) * av.y;
            float e2 = __expf(c2 - mrow[i]) * av.z;
            float e3 = __expf(c3 - mrow[i]) * av.w;
            dp[i] += (e0 + e1) + (e2 + e3);
        }
    }
    float invd[16];
#pragma unroll
    for (int i = 0; i < 16; ++i) {
        float s = wave_sum32(dp[i]);
        invd[i] = (s == 0.f) ? 1.f : (1.f / s);
    }

    // ---- pass 2: write normalized att + accumulate hp with WMMA ----
    v8f acc[8];
#pragma unroll
    for (int dt = 0; dt < 8; ++dt)
#pragma unroll
        for (int v = 0; v < 8; ++v) acc[dt][v] = 0.f;

    float* e = smem + hd * (16 * ESTRIDE);
    float* attB = att + ((size_t)(b * H_ + hd) * N_ + n0) * N_;

#pragma unroll 1
    for (int j0 = 0; j0 < N_; j0 += JC_) {
        float4 s2v = *(const float4*)(s2p + j0 + 4 * lane);
#pragma unroll
        for (int i = 0; i < 16; ++i) {
            float4 av = *(const float4*)(adjB + (size_t)i * N_ + j0 + 4 * lane);
            float c0 = s1row[i] + s2v.x; c0 = c0 > 0.f ? c0 : ALPHA_ * c0;
            float c1 = s1row[i] + s2v.y; c1 = c1 > 0.f ? c1 : ALPHA_ * c1;
            float c2 = s1row[i] + s2v.z; c2 = c2 > 0.f ? c2 : ALPHA_ * c2;
            float c3 = s1row[i] + s2v.w; c3 = c3 > 0.f ? c3 : ALPHA_ * c3;
            float4 o;
            o.x = __expf(c0 - mrow[i]) * av.x * invd[i];
            o.y = __expf(c1 - mrow[i]) * av.y * invd[i];
            o.z = __expf(c2 - mrow[i]) * av.z * invd[i];
            o.w = __expf(c3 - mrow[i]) * av.w * invd[i];
            *(float4*)(attB + (size_t)i * N_ + j0 + 4 * lane) = o;   // output att
            *(float4*)(e + i * ESTRIDE + 4 * lane) = o;              // LDS stage
        }

        // hp_tile(16xD) += att_chunk(16xJC) @ ht_chunk(JCxD), f32 WMMA
        const float* eA   = e + lo * ESTRIDE + 2 * hi;               // A frag base
        const float* hrow = htp + (size_t)(j0 + 2 * hi) * D_ + lo;   // B frag base
#pragma unroll 2
        for (int k0 = 0; k0 < JC_; k0 += 4) {
            v2f afrag = *(const v2f*)(eA + k0);                      // ds_load_b64
            const float* hr = hrow + (size_t)k0 * D_;
#pragma unroll
            for (int dt = 0; dt < 8; ++dt) {
                v2f bfrag;
                bfrag.x = hr[dt * 16];
                bfrag.y = hr[dt * 16 + D_];
                acc[dt] = __builtin_amdgcn_wmma_f32_16x16x4_f32(
                    false, afrag, false, bfrag, (short)0, acc[dt], false, false);
            }
        }
    }

    // ---- fused epilogue: mean over heads + tanh ----
    __syncthreads();                       // waves done with e regions
    float* hb = smem + hd * (16 * 128);    // reuse staging memory
#pragma unroll
    for (int dt = 0; dt < 8; ++dt)
#pragma unroll
        for (int v = 0; v < 8; ++v)
            hb[(size_t)(v + 8 * hi) * 128 + dt * 16 + lo] = acc[dt][v];
    __syncthreads();

    const float* xb = x + ((size_t)b * N_ + n0) * D_;
    float* ob = out + ((size_t)b * N_ + n0) * D_;
    for (int idx = threadIdx.x; idx < 16 * 128; idx += 128) {
        float s = smem[idx] + smem[2048 + idx] + smem[4096 + idx] + smem[6144 + idx];
        ob[idx] = tanhf(xb[idx] + 0.25f * s);
    }
}

// ---------------------------------------------------------------------------
extern "C" void kernel_launch(void* const* d_in, const int* in_sizes, int n_in,
                              void* d_out, int out_size, void* d_ws, size_t ws_size,
                              hipStream_t stream)
{
    const float* h   = (const float*)d_in[0];   // (B,N,1,D) == (B,N,D)
    const float* adj = (const float*)d_in[1];   // (B,N,N)
    const float* W   = (const float*)d_in[2];   // (H,D,D)
    const float* Wb  = (const float*)d_in[3];   // (H,D)
    const float* a1  = (const float*)d_in[4];   // (H,D)
    const float* a1b = (const float*)d_in[5];   // (H,)
    const float* a2  = (const float*)d_in[6];   // (H,D)
    const float* a2b = (const float*)d_in[7];   // (H,)

    float* ws  = (float*)d_ws;
    float* ht  = ws;                                        // B*H*N*D
    float* s1  = ht + (size_t)B_ * H_ * N_ * D_;            // B*H*N
    float* s2  = s1 + (size_t)B_ * H_ * N_;                 // B*H*N
    float* s2m = s2 + (size_t)B_ * H_ * N_;                 // B*H

    float* out = (float*)d_out;                             // B*N*D
    float* att = out + (size_t)B_ * N_ * D_;                // B*H*N*N

    k_ht<<<B_ * H_ * NT_, 32, 0, stream>>>(h, W, Wb, a1, a1b, a2, a2b, ht, s1, s2);
    k_s2max<<<1, 512, 0, stream>>>(s2, s2m);
    k_att<<<B_ * NT_, 128, 0, stream>>>(adj, h, ht, s1, s2, s2m, out, att);
}